// SparseLinear_26448408609383
// MI455X (gfx1250) — compile-verified
//
#include <hip/hip_runtime.h>

// Padded-CSR SparseLinear  ->  densify W + fp32 WMMA GEMM
// y[B=8192, OUT=4096] = x[8192, 4096] @ Wdense[4096, 4096] + bias
#define IN_F   4096
#define OUT_F  4096
#define BATCH  8192
#define CAP    492

typedef __attribute__((ext_vector_type(2))) float v2f;
typedef __attribute__((ext_vector_type(8))) float v8f;

// ---------------------------------------------------------------------------
// Kernel 1: zero the dense weight matrix (64 MB in d_ws)
// ---------------------------------------------------------------------------
__global__ __launch_bounds__(256) void zero_w_kernel(float4* __restrict__ w4) {
    size_t i = (size_t)blockIdx.x * blockDim.x + threadIdx.x;  // exactly sized grid
    w4[i] = make_float4(0.f, 0.f, 0.f, 0.f);
}

// ---------------------------------------------------------------------------
// Kernel 2: deterministic scatter of padded-CSR rows into dense W[k][n].
// Thread o owns dense column o (addresses Wd[c*OUT_F + o]); it is the ONLY
// writer of that column, so duplicate col_idx entries are accumulated
// serially in a fixed order -> bitwise deterministic, no atomics needed.
// ---------------------------------------------------------------------------
__global__ __launch_bounds__(256) void scatter_w_kernel(
    const float* __restrict__ values, const int* __restrict__ col_idx,
    float* __restrict__ wd) {
    const int o = blockIdx.x * blockDim.x + threadIdx.x;      // 0..OUT_F-1
    const float* vrow = values  + (size_t)o * CAP;
    const int*   crow = col_idx + (size_t)o * CAP;
    for (int k = 0; k < CAP; ++k) {
        const float v = vrow[k];
        if (v != 0.0f) {                     // skip padding slots (value==0)
            const int c = crow[k];
            wd[(size_t)c * OUT_F + o] += v;  // single-owner RMW, deterministic
        }
    }
}

// ---------------------------------------------------------------------------
// Kernel 3: fp32 GEMM via V_WMMA_F32_16X16X4_F32.
// Workgroup tile 128x128, BK=16; 8 waves, each wave = 32(M) x 64(N)
// = 2x4 accumulators. Register double-buffered global->LDS staging.
// ---------------------------------------------------------------------------
#define BM  128
#define BN  128
#define BK  16
#define LDP 18   // padded LDS row stride (floats): conflict-free frag reads

__global__ __launch_bounds__(256) void gemm_wmma_f32_kernel(
    const float* __restrict__ X,    // [BATCH, IN_F] row-major
    const float* __restrict__ W,    // [IN_F, OUT_F] row-major (K-major)
    const float* __restrict__ bias, // [OUT_F]
    float* __restrict__ Y) {        // [BATCH, OUT_F]
    __shared__ float Xs[BM][LDP];   // Xs[m][k]
    __shared__ float Ws[BN][LDP];   // Ws[n][k]  (transposed chunk of W)

    const int tid  = threadIdx.x;
    const int lane = tid & 31;
    const int wv   = tid >> 5;            // wave id 0..7
    const int wm   = (wv & 3) * 32;       // wave M offset in block tile
    const int wn   = (wv >> 2) * 64;      // wave N offset in block tile
    const int m0   = blockIdx.y * BM;
    const int n0   = blockIdx.x * BN;
    const int lm   = lane & 15;           // row/col within 16-wide fragment
    const int kb   = (lane >> 4) << 1;    // K pair base: 0 or 2 (A/B layout)

    // global->LDS staging coordinates (256 threads move 128x16 + 16x128 fp32)
    const int xr = tid >> 1;              // 0..127 : X row within tile
    const int xc = (tid & 1) * 8;         // 0 or 8 : X col (k) within chunk
    const int wr = tid >> 4;              // 0..15  : W row (k) within chunk
    const int wc = (tid & 15) * 8;        // 0..120 : W col (n) within tile

    const float* xg = X + (size_t)(m0 + xr) * IN_F + xc;
    const float* wg = W + (size_t)wr * OUT_F + (n0 + wc);

    v8f acc[2][4];
#pragma unroll
    for (int t = 0; t < 2; ++t)
#pragma unroll
        for (int j = 0; j < 4; ++j) acc[t][j] = 0.f;

    // prologue: stage first K-chunk into registers
    float4 xa0 = *(const float4*)(xg + 0);
    float4 xa1 = *(const float4*)(xg + 4);
    float4 wa0 = *(const float4*)(wg + 0);
    float4 wa1 = *(const float4*)(wg + 4);

    for (int k0 = 0; k0 < IN_F; k0 += BK) {
        __syncthreads();  // previous iteration's LDS reads are done
        *(float4*)&Xs[xr][xc]     = xa0;
        *(float4*)&Xs[xr][xc + 4] = xa1;
#pragma unroll
        for (int i = 0; i < 4; ++i) Ws[wc + i][wr]     = (&wa0.x)[i];
#pragma unroll
        for (int i = 0; i < 4; ++i) Ws[wc + 4 + i][wr] = (&wa1.x)[i];
        __syncthreads();

        // overlap: fetch next chunk into registers while computing this one
        if (k0 + BK < IN_F) {
            xa0 = *(const float4*)(xg + (k0 + BK));
            xa1 = *(const float4*)(xg + (k0 + BK) + 4);
            wa0 = *(const float4*)(wg + (size_t)(k0 + BK) * OUT_F);
            wa1 = *(const float4*)(wg + (size_t)(k0 + BK) * OUT_F + 4);
        }

#pragma unroll
        for (int ks = 0; ks < 4; ++ks) {          // 4 WMMA K-steps of 4
            const int kk = ks * 4 + kb;
            // A fragments (16x4 fp32): lane -> row lm, K pair kb
            v2f a0 = *(const v2f*)&Xs[wm + lm][kk];
            v2f a1 = *(const v2f*)&Xs[wm + 16 + lm][kk];
            // B fragments (4x16 fp32): lane -> col lm, K pair kb
            v2f b0 = *(const v2f*)&Ws[wn + lm][kk];
            v2f b1 = *(const v2f*)&Ws[wn + 16 + lm][kk];
            v2f b2 = *(const v2f*)&Ws[wn + 32 + lm][kk];
            v2f b3 = *(const v2f*)&Ws[wn + 48 + lm][kk];

            acc[0][0] = __builtin_amdgcn_wmma_f32_16x16x4_f32(
                false, a0, false, b0, (short)0, acc[0][0], false, false);
            acc[0][1] = __builtin_amdgcn_wmma_f32_16x16x4_f32(
                false, a0, false, b1, (short)0, acc[0][1], false, false);
            acc[0][2] = __builtin_amdgcn_wmma_f32_16x16x4_f32(
                false, a0, false, b2, (short)0, acc[0][2], false, false);
            acc[0][3] = __builtin_amdgcn_wmma_f32_16x16x4_f32(
                false, a0, false, b3, (short)0, acc[0][3], false, false);
            acc[1][0] = __builtin_amdgcn_wmma_f32_16x16x4_f32(
                false, a1, false, b0, (short)0, acc[1][0], false, false);
            acc[1][1] = __builtin_amdgcn_wmma_f32_16x16x4_f32(
                false, a1, false, b1, (short)0, acc[1][1], false, false);
            acc[1][2] = __builtin_amdgcn_wmma_f32_16x16x4_f32(
                false, a1, false, b2, (short)0, acc[1][2], false, false);
            acc[1][3] = __builtin_amdgcn_wmma_f32_16x16x4_f32(
                false, a1, false, b3, (short)0, acc[1][3], false, false);
        }
    }

    // epilogue: C/D layout -> VGPR r holds M=r (lanes 0-15) / M=r+8 (16-31),
    // N = lane&15. Add bias (function of N only) and store fp32.
#pragma unroll
    for (int j = 0; j < 4; ++j) {
        const int n  = n0 + wn + j * 16 + lm;
        const float bv = bias[n];
#pragma unroll
        for (int t = 0; t < 2; ++t) {
            const int mbase = m0 + wm + t * 16 + ((lane >> 4) << 3);
#pragma unroll
            for (int r = 0; r < 8; ++r) {
                Y[(size_t)(mbase + r) * OUT_F + n] = acc[t][j][r] + bv;
            }
        }
    }
}

// ---------------------------------------------------------------------------
extern "C" void kernel_launch(void* const* d_in, const int* in_sizes, int n_in,
                              void* d_out, int out_size, void* d_ws, size_t ws_size,
                              hipStream_t stream) {
    const float* x       = (const float*)d_in[0];  // [8192, 4096]
    const float* values  = (const float*)d_in[1];  // [4096, 492]
    const int*   col_idx = (const int*)d_in[2];    // [4096, 492]
    const float* bias    = (const float*)d_in[3];  // [4096]
    float* y  = (float*)d_out;                     // [8192, 4096]
    float* wd = (float*)d_ws;                      // dense W: 4096*4096 fp32 = 64 MB

    // 1) W_dense = 0
    {
        const size_t n4 = (size_t)IN_F * OUT_F / 4;           // float4 count
        zero_w_kernel<<<dim3((unsigned)(n4 / 256)), dim3(256), 0, stream>>>(
            (float4*)wd);
    }
    // 2) scatter padded-CSR into W_dense (one thread per output column)
    scatter_w_kernel<<<dim3(OUT_F / 256), dim3(256), 0, stream>>>(
        values, col_idx, wd);
    // 3) y = x @ W_dense + bias   (fp32 WMMA GEMM)
    gemm_wmma_f32_kernel<<<dim3(OUT_F / BN, BATCH / BM), dim3(256), 0, stream>>>(
        x, wd, bias, y);
}